// DepthEmissionRaymarcher_88656714925050
// MI455X (gfx1250) — compile-verified
//
#include <hip/hip_runtime.h>
#include <stdint.h>

// DepthEmissionRaymarcher for MI455X (gfx1250, wave32).
// One wave per ray. Memory-bound (AI ~0.5 flop/byte): optimize data movement,
// not math. Feature tile (8KB/ray) is streamed global->LDS with the CDNA5
// async-to-LDS path (ASYNCcnt), overlapped with the density prefix-scan.
// All once-consumed streams use NT temporal hints (stream >> 192MB L2).

#define WAVES_PER_BLOCK 4
#define RM_P 64
#define RM_F 32
#define FEAT_FLOATS (RM_P * RM_F)   // 2048 floats = 8KB per ray

#if __has_builtin(__builtin_amdgcn_global_load_async_to_lds_b128)
#define HAVE_ASYNC_LDS 1
#else
#define HAVE_ASYNC_LDS 0
#endif

// gfx12-style CPol: TH[2:0]; TH=1 -> non-temporal (ISA 7.3 Load TH "NT")
#define CPOL_NT 1

typedef float v4f __attribute__((ext_vector_type(4)));   // clang-native float4

#if defined(__AMDGCN__)
typedef int v4i __attribute__((ext_vector_type(4)));
typedef __attribute__((address_space(1))) v4i gas_v4i;   // global (AS1) int4
typedef __attribute__((address_space(3))) v4i las_v4i;   // LDS    (AS3) int4

__device__ __forceinline__ gas_v4i* to_global_v4(const void* p) {
    v4i* g = (v4i*)p;            // bitcast in generic AS
    return (gas_v4i*)g;          // addrspacecast generic -> AS1
}
__device__ __forceinline__ las_v4i* to_lds_v4(void* p) {
    v4i* g = (v4i*)p;            // bitcast in generic AS
    return (las_v4i*)g;          // addrspacecast generic -> AS3
}
#endif

__global__ __launch_bounds__(WAVES_PER_BLOCK * 32)
void raymarch_kernel(const float* __restrict__ dens,
                     const float* __restrict__ feat,
                     const float* __restrict__ org,
                     const float* __restrict__ dir,
                     const float* __restrict__ len,
                     float* __restrict__ outDepth,
                     float* __restrict__ outFeat,
                     int nRays)
{
    __shared__ __align__(16) float sFeat[WAVES_PER_BLOCK][FEAT_FLOATS];
    __shared__ float sProb[WAVES_PER_BLOCK][RM_P];

    const int lane = threadIdx.x & 31;
    const int wave = threadIdx.x >> 5;
    const int ray  = blockIdx.x * WAVES_PER_BLOCK + wave;
    if (ray >= nRays) return;   // wave-uniform

    // ---------------- stage 1: kick off async feature streaming ------------
    const float* gsrc = feat + (size_t)ray * FEAT_FLOATS;
#if HAVE_ASYNC_LDS
    // 16 x (32 lanes x 16B) = 8KB, lands directly in LDS, tracked by ASYNCcnt.
    // NT policy: stream is 671MB, consumed once -> don't thrash L2.
#pragma unroll
    for (int j = 0; j < 16; ++j) {
        const float* gp = gsrc + j * 128 + lane * 4;
        float*       lp = &sFeat[wave][j * 128 + lane * 4];
        __builtin_amdgcn_global_load_async_to_lds_b128(
            to_global_v4(gp), to_lds_v4(lp), 0, CPOL_NT);
    }
#else
    // fallback: synchronous 128-bit copies through VGPRs
#pragma unroll
    for (int j = 0; j < 16; ++j) {
        *reinterpret_cast<v4f*>(&sFeat[wave][j * 128 + lane * 4]) =
            __builtin_nontemporal_load(
                reinterpret_cast<const v4f*>(gsrc + j * 128 + lane * 4));
    }
#endif

    // ---------------- stage 2: probs (overlaps with the async DMA) ---------
    // floor-plane depth: fd = -oz/dz; nan -> -1; <=0 -> 1e4
    float oz = org[(size_t)ray * 3 + 2];
    float dz = dir[(size_t)ray * 3 + 2];
    float fd = (0.0f - oz) / dz;
    if (fd != fd) fd = -1.0f;          // NaN (0/0)
    if (fd <= 0.0f) fd = 1e4f;         // includes -inf path

    const float* lp64 = len  + (size_t)ray * RM_P;
    const float* dp64 = dens + (size_t)ray * RM_P;
    float lenLo = __builtin_nontemporal_load(lp64 + lane);
    float lenHi = __builtin_nontemporal_load(lp64 + lane + 32);
    float dLo   = __builtin_nontemporal_load(dp64 + lane);
    float dHi   = __builtin_nontemporal_load(dp64 + lane + 32);
    if (lane == 31) dHi = 1.0f;        // last sample opaque
    if (lenLo > fd) dLo = 1.0f;        // below floor -> opaque
    if (lenHi > fd) dHi = 1.0f;

    // inclusive prefix sum over 64 samples = two wave32 scans
    float sLo = dLo;
#pragma unroll
    for (int off = 1; off < 32; off <<= 1) {
        float t = __shfl_up(sLo, off, 32);
        if (lane >= off) sLo += t;
    }
    float tot = __shfl(sLo, 31, 32);
    float sHi = dHi;
#pragma unroll
    for (int off = 1; off < 32; off <<= 1) {
        float t = __shfl_up(sHi, off, 32);
        if (lane >= off) sHi += t;
    }
    sHi += tot;

    // clamp + adjacent difference -> per-sample hit probability
    float mLo  = fminf(sLo, 1.0f);
    float mHi  = fminf(sHi, 1.0f);
    float mTot = fminf(tot, 1.0f);                 // clamped cumsum at p=31
    float pmLo = __shfl_up(mLo, 1, 32); if (lane == 0) pmLo = 0.0f;
    float pmHi = __shfl_up(mHi, 1, 32); if (lane == 0) pmHi = mTot;
    float probLo = mLo - pmLo;
    float probHi = mHi - pmHi;

    sProb[wave][lane]      = probLo;
    sProb[wave][lane + 32] = probHi;

    // depth = sum(probs * lengths): wave32 butterfly reduction
    float dep = probLo * lenLo + probHi * lenHi;
#pragma unroll
    for (int off = 16; off; off >>= 1) dep += __shfl_xor(dep, off, 32);
    if (lane == 0) __builtin_nontemporal_store(dep, outDepth + ray);

    // ---------------- stage 3: wait for DMA, accumulate features -----------
#if HAVE_ASYNC_LDS
#if __has_builtin(__builtin_amdgcn_s_wait_asynccnt)
    __builtin_amdgcn_s_wait_asynccnt(0);
#else
    asm volatile("s_wait_asynccnt 0" ::: "memory");
#endif
    asm volatile("" ::: "memory");  // keep ds reads below the wait
#endif

    // lane l owns features (l%8)*4..+3 of sample 4*i + l/8 in iteration i
    const int sub = lane >> 3;
    v4f acc = (v4f)0.0f;
#pragma unroll
    for (int i = 0; i < 16; ++i) {
        v4f   v  = *reinterpret_cast<const v4f*>(&sFeat[wave][i * 128 + lane * 4]);
        float pr = sProb[wave][i * 4 + sub];       // LDS broadcast read
        acc += v * pr;                             // 4x v_fma per lane
    }
    // combine the 4 lane-groups (xor-8, xor-16); then lanes 0..7 hold F=0..31
#pragma unroll
    for (int off = 8; off <= 16; off <<= 1) {
        acc.x += __shfl_xor(acc.x, off, 32);
        acc.y += __shfl_xor(acc.y, off, 32);
        acc.z += __shfl_xor(acc.z, off, 32);
        acc.w += __shfl_xor(acc.w, off, 32);
    }
    if (lane < 8) {
        __builtin_nontemporal_store(
            acc, reinterpret_cast<v4f*>(&outFeat[(size_t)ray * RM_F + lane * 4]));
    }
}

extern "C" void kernel_launch(void* const* d_in, const int* in_sizes, int n_in,
                              void* d_out, int out_size, void* d_ws, size_t ws_size,
                              hipStream_t stream) {
    // setup_inputs order: rays_densities, rays_features, origins, directions, lengths
    const float* dens = (const float*)d_in[0];
    const float* feat = (const float*)d_in[1];
    const float* org  = (const float*)d_in[2];
    const float* dir  = (const float*)d_in[3];
    const float* len  = (const float*)d_in[4];

    const int nRays = in_sizes[2] / 3;             // B*H*W = 81920

    // outputs concatenated: depth (nRays) then features (nRays*F)
    float* outDepth = (float*)d_out;
    float* outFeat  = outDepth + nRays;

    const int blocks = (nRays + WAVES_PER_BLOCK - 1) / WAVES_PER_BLOCK;
    raymarch_kernel<<<blocks, WAVES_PER_BLOCK * 32, 0, stream>>>(
        dens, feat, org, dir, len, outDepth, outFeat, nRays);
}